// loss_con1_22591527977071
// MI455X (gfx1250) — compile-verified
//
#include <hip/hip_runtime.h>
#include <hip/hip_bf16.h>

typedef __attribute__((ext_vector_type(2))) float v2f;
typedef __attribute__((ext_vector_type(8))) float v8f;

#define N_ELEMS         8192
#define THREADS         256
#define WAVES_PER_BLOCK (THREADS / 32)
#define ELEMS_PER_WAVE  (N_ELEMS / WAVES_PER_BLOCK)   // 1024
#define ITERS           (ELEMS_PER_WAVE / 64)         // 16 WMMA pairs per wave

// One block per input vector. Computes S = sum(v), Q = sum(v*v) using
// V_WMMA_F32_16X16X4_F32 with an all-ones B matrix as the reduction engine:
// D[m][n] = sum_k A[m][k] + C[m][n]. Summation is order-invariant, so any
// bijection of the 64 loaded elements into A's register slots is valid, and
// summing every C component over every lane yields 16x the true sum
// (independent of the exact VGPR layout).
__global__ __launch_bounds__(THREADS)
void wmma_reduce_sums_kernel(const float* __restrict__ v0,
                             const float* __restrict__ v1,
                             const float* __restrict__ v2,
                             const float* __restrict__ v3,
                             float* __restrict__ ws) {
    __shared__ float acc[2];
    const int tid = threadIdx.x;
    if (tid == 0) { acc[0] = 0.0f; acc[1] = 0.0f; }
    __syncthreads();

    const float* src;
    switch (blockIdx.x) {
        case 0:  src = v0; break;
        case 1:  src = v1; break;
        case 2:  src = v2; break;
        default: src = v3; break;
    }

    const int wave = tid >> 5;
    const int lane = tid & 31;

    v8f cs = {};                       // accumulates sum(x)
    v8f cq = {};                       // accumulates sum(x*x)
    const v2f ones = {1.0f, 1.0f};

    const float2* src2 = (const float2*)src;
    // Per iteration: this wave consumes 64 consecutive floats (32 lanes x float2).
    // Loop bounds are wave-uniform -> EXEC is all-1s at every WMMA (ISA req).
    for (int t = 0; t < ITERS; ++t) {
        float2 d = src2[wave * (ELEMS_PER_WAVE / 2) + t * 32 + lane];
        v2f a;  a.x  = d.x;       a.y  = d.y;
        v2f a2; a2.x = d.x * d.x; a2.y = d.y * d.y;
        // 8 args: (neg_a, A, neg_b, B, c_mod, C, reuse_a, reuse_b)
        cs = __builtin_amdgcn_wmma_f32_16x16x4_f32(false, a,  false, ones,
                                                   (short)0, cs, false, false);
        cq = __builtin_amdgcn_wmma_f32_16x16x4_f32(false, a2, false, ones,
                                                   (short)0, cq, false, false);
    }

    // Sum of all C components across all lanes == sum over the full 16x16 D
    // matrix == 16 * wave_total (every column of D holds the wave total).
    float s = cs[0] + cs[1] + cs[2] + cs[3] + cs[4] + cs[5] + cs[6] + cs[7];
    float q = cq[0] + cq[1] + cq[2] + cq[3] + cq[4] + cq[5] + cq[6] + cq[7];

    atomicAdd(&acc[0], s);   // ds_add_f32
    atomicAdd(&acc[1], q);
    __syncthreads();

    if (tid == 0) {
        ws[2 * blockIdx.x + 0] = acc[0] * 0.0625f;  // /16, exact
        ws[2 * blockIdx.x + 1] = acc[1] * 0.0625f;
    }
}

// Closed form:
//   mean_{i,j}(1 + x_i - y_j)^2
//     = 1 + Q_x/N + Q_y/N + 2*S_x/N - 2*S_y/N - 2*(S_x/N)*(S_y/N)
__global__ void finalize_loss_kernel(const float* __restrict__ ws,
                                     const float* __restrict__ lam,
                                     float* __restrict__ out) {
    if (threadIdx.x == 0 && blockIdx.x == 0) {
        const float Ninv = 1.0f / (float)N_ELEMS;
        float Syp = ws[0], Qyp = ws[1];   // y_pred_pos
        float Syn = ws[2], Qyn = ws[3];   // y_pred_neg
        float Sgp = ws[4], Qgp = ws[5];   // g_pred_pos
        float Sgn = ws[6], Qgn = ws[7];   // g_pred_neg

        float mx, my;
        // loss1 = L(y_neg, y_pos)
        mx = Syn * Ninv; my = Syp * Ninv;
        float l1 = 1.0f + Qyn * Ninv + Qyp * Ninv + 2.0f * mx - 2.0f * my - 2.0f * mx * my;
        // loss2 = L(g_pos, g_neg)
        mx = Sgp * Ninv; my = Sgn * Ninv;
        float l2 = 1.0f + Qgp * Ninv + Qgn * Ninv + 2.0f * mx - 2.0f * my - 2.0f * mx * my;
        // loss3 = L(g_neg, g_pos)
        mx = Sgn * Ninv; my = Sgp * Ninv;
        float l3 = 1.0f + Qgn * Ninv + Qgp * Ninv + 2.0f * mx - 2.0f * my - 2.0f * mx * my;

        out[0] = lam[0] * l1 + lam[1] * l2 + lam[2] * l3;
    }
}

extern "C" void kernel_launch(void* const* d_in, const int* in_sizes, int n_in,
                              void* d_out, int out_size, void* d_ws, size_t ws_size,
                              hipStream_t stream) {
    const float* y_pred_pos = (const float*)d_in[0];
    const float* y_pred_neg = (const float*)d_in[1];
    const float* g_pred_pos = (const float*)d_in[2];
    const float* g_pred_neg = (const float*)d_in[3];
    const float* lam        = (const float*)d_in[4];
    float* ws  = (float*)d_ws;   // 8 floats: {S,Q} per vector
    float* out = (float*)d_out;

    wmma_reduce_sums_kernel<<<dim3(4), dim3(THREADS), 0, stream>>>(
        y_pred_pos, y_pred_neg, g_pred_pos, g_pred_neg, ws);
    finalize_loss_kernel<<<dim3(1), dim3(32), 0, stream>>>(ws, lam, out);
}